// PermLayer_72189810312026
// MI455X (gfx1250) — compile-verified
//
#include <hip/hip_runtime.h>
#include <stdint.h>

// PermLayer for MI455X (gfx1250): pure shuffle, memory-bound.
//   traffic: 32 MB read (once, async-staged to LDS) + 256 MB written (once,
//   non-temporal) -> ~12.4 us floor at 23.3 TB/s. Zero required FLOPs, so no
//   WMMA; CDNA5 paths used: global_load_async_to_lds_b128 (ASYNCcnt),
//   s_wait_asynccnt, split barriers, ds_load_b64 gathers, NT b128 stores.
// Index generation: keyed 4-round Feistel bijection on [0,256) per
//   (row, perm, group) - replaces the reference's argsort+index-tensor with
//   O(1) in-register index math (same construct as the reference: an
//   independent pseudorandom permutation per group; not bit-identical to
//   JAX threefry+argsort).
// Work mapping: wave w of 8 owns pairs [w*128, w*128+128) => group g = w>>1
//   is WAVE-UNIFORM, so all key derivation runs on the scalar unit
//   (readfirstlane-scalarized); only the 4 VALU ops/round of the Feistel
//   mix stay vector.

#define B_ROWS 4096
#define P_PERM 8
#define D_COLS 2048

typedef float v4f __attribute__((ext_vector_type(4)));

// Keyed bijection on [0,256): 4-round balanced Feistel over 4+4 bit halves.
// `key` is wave-uniform (SGPR); per-round constants fold into it on SALU.
__device__ __forceinline__ uint32_t perm256(uint32_t i, uint32_t key) {
  uint32_t l = i & 0xFu;
  uint32_t r = (i >> 4) & 0xFu;
#pragma unroll
  for (uint32_t rnd = 0; rnd < 4; ++rnd) {
    uint32_t f = ((r ^ (key + rnd * 0x632BE5ABu)) * 0x9E3779B9u) >> 28;
    uint32_t nl = r;
    r = l ^ f;
    l = nl;
  }
  return (r << 4) | l;
}

__global__ __launch_bounds__(256) void perm_layer_kernel(
    const float* __restrict__ x, float* __restrict__ out) {
  __shared__ float row[D_COLS];  // 8 KB of the 320 KB/WGP LDS
  const int b = blockIdx.x;
  const int t = threadIdx.x;  // 0..255 (8 wave32s)

  // ---- Stage row b into LDS via CDNA5 async global->LDS DMA path ----
  // Each lane copies 32 B: one b128 at byte t*16, one at t*16+4096.
  // INST_OFFSET is added to BOTH the LDS and global addresses, so one base
  // address pair serves both halves.
  {
    unsigned lds_off = (unsigned)(size_t)(void*)&row[0] + (unsigned)t * 16u;
    uint64_t gaddr =
        (uint64_t)(size_t)(const void*)(x + (size_t)b * D_COLS) +
        (uint64_t)t * 16u;
    asm volatile("global_load_async_to_lds_b128 %0, %1, off"
                 :
                 : "v"(lds_off), "v"(gaddr)
                 : "memory");
    asm volatile("global_load_async_to_lds_b128 %0, %1, off offset:4096"
                 :
                 : "v"(lds_off), "v"(gaddr)
                 : "memory");
    asm volatile("s_wait_asynccnt 0" ::: "memory");
  }
  __syncthreads();

  const float2* pairs = (const float2*)row;

  // Wave-uniform work mapping: wave w owns pairs [w*128, w*128+128),
  // all inside group g = w>>1. readfirstlane proves uniformity to the
  // compiler so the key chain lives in SGPRs.
  const uint32_t w =
      (uint32_t)__builtin_amdgcn_readfirstlane((int)((uint32_t)t >> 5));
  const uint32_t lane = (uint32_t)t & 31u;
  const uint32_t g = w >> 1;               // group 0..3, wave-uniform
  const uint32_t gbase = g << 8;           // first pair index of the group
  const uint32_t h0a = w * 128u + 2u * lane;  // pairs h0a, h0a+1
  const uint32_t h0b = h0a + 64u;             // pairs h0b, h0b+1 (same group)
  const uint32_t key_bg =
      ((((uint32_t)b << 5) ^ 0x5bf03635u) | g);  // scalar per (block, wave)

  for (int p = 0; p < P_PERM; ++p) {
    v4f* orow = (v4f*)(out + ((size_t)b * P_PERM + (size_t)p) * D_COLS);
    // Fully scalar key derivation per (wave, perm).
    const uint32_t key = (uint32_t)__builtin_amdgcn_readfirstlane(
        (int)((key_bg | ((uint32_t)p << 2)) * 0x85EBCA6Bu));

#pragma unroll
    for (int half = 0; half < 2; ++half) {
      const uint32_t h0 = half ? h0b : h0a;
      const uint32_t loc = h0 & 255u;
      const uint32_t s0 = gbase | perm256(loc, key);
      const uint32_t s1 = gbase | perm256(loc + 1u, key);
      const float2 a = pairs[s0];  // ds_load_b64 gather
      const float2 c = pairs[s1];  // ds_load_b64 gather
      v4f v = {a.x, a.y, c.x, c.y};
      // Write-once 256 MB stream: non-temporal b128 store (th:TH_STORE_NT).
      __builtin_nontemporal_store(v, orow + (h0 >> 1));
    }
  }
}

extern "C" void kernel_launch(void* const* d_in, const int* in_sizes, int n_in,
                              void* d_out, int out_size, void* d_ws,
                              size_t ws_size, hipStream_t stream) {
  (void)in_sizes; (void)n_in; (void)out_size; (void)d_ws; (void)ws_size;
  const float* x = (const float*)d_in[0];
  float* out = (float*)d_out;
  perm_layer_kernel<<<dim3(B_ROWS), dim3(256), 0, stream>>>(x, out);
}